// CharModel_56659208569192
// MI455X (gfx1250) — compile-verified
//
#include <hip/hip_runtime.h>

// Problem dims (fixed by the reference).
#define B_    32
#define S_    1024
#define W_    512
#define D_    768
#define NPOS_ 32

#define TPB   192   // D/4 float4 lanes -> one full 768-float row per step, 6 wave32s
#define CHUNK 8     // rows staged per async round (8 * 3KB = 24KB LDS)

typedef float float4v __attribute__((ext_vector_type(4)));

// gfx1250 async global->LDS path, guarded so the file compiles on any toolchain.
#if defined(__AMDGCN__) && defined(__gfx1250__) && \
    __has_builtin(__builtin_amdgcn_global_load_async_to_lds_b128) && \
    __has_builtin(__builtin_amdgcn_s_wait_asynccnt)
#define USE_ASYNC 1
// Exact pointee type from the compiler diagnostic: vector_size(16) int.
typedef int v4i __attribute__((vector_size(16)));
typedef __attribute__((address_space(1))) v4i* gas_v4i_t;   // global (prints as __device__)
typedef __attribute__((address_space(3))) v4i* lds_v4i_t;   // LDS
#else
#define USE_ASYNC 0
#endif

// One block per (batch, word). Segment mean over feats rows + pos embedding add.
__global__ __launch_bounds__(TPB) void seg_mean_pos_kernel(
    const float* __restrict__ feats,      // [B, S, D] f32
    const int*   __restrict__ word_lens,  // [B, W] i32 (segment starts, 0 = pad except j==0)
    const int*   __restrict__ seq_len,    // [B] i32
    const int*   __restrict__ pos,        // [B, W] i32
    const float* __restrict__ pos_table,  // [NPOS, D] f32 (row 0 is zeros)
    float*       __restrict__ out)        // [B, W, D] f32
{
#if USE_ASYNC
    __shared__ float4v stage[CHUNK * TPB];   // 24 KB of the 320 KB WGP LDS
#endif
    const int j = blockIdx.x;     // word index   [0, W)
    const int b = blockIdx.y;     // batch index  [0, B)
    const int t = threadIdx.x;    // float4 lane  [0, 192)

    const int  base  = b * W_ + j;
    const int  start = word_lens[base];
    const int  nxt   = (j + 1 < W_) ? word_lens[base + 1] : 0;
    const int  end   = (nxt == 0) ? seq_len[b] : nxt;
    const bool valid = (start != 0) || (j == 0);
    const int  p     = pos[base];

    float4v acc = {0.0f, 0.0f, 0.0f, 0.0f};

    if (valid) {
        int len = end - start;
        if (len < 1) len = 1;                 // matches jnp.maximum(ends - starts, 1)
        const int stride4 = D_ / 4;           // float4s per char row
        const float4v* __restrict__ row =
            (const float4v*)(feats + ((size_t)b * S_ + start) * (size_t)D_) + t;

        int c = 0;
#if USE_ASYNC
        // Stage up to CHUNK rows per round on ASYNCcnt, then sum from LDS.
        // Each lane reads back only its own 16B slot -> no barrier required,
        // s_wait_asynccnt 0 orders this wave's async writes vs. its DS reads.
        while (c < len) {
            const int n = (len - c < CHUNK) ? (len - c) : CHUNK;
#pragma unroll
            for (int r = 0; r < CHUNK; ++r) {
                if (r < n) {
                    float4v* src_nc = const_cast<float4v*>(row + (size_t)(c + r) * stride4);
                    __builtin_amdgcn_global_load_async_to_lds_b128(
                        (gas_v4i_t)src_nc,
                        (lds_v4i_t)&stage[r * TPB + t],
                        /*offset=*/0, /*cpol=*/0);
                }
            }
            __builtin_amdgcn_s_wait_asynccnt(0);
#pragma unroll
            for (int r = 0; r < CHUNK; ++r) {
                if (r < n) acc += stage[r * TPB + t];
            }
            c += n;
        }
#else
        // Streaming fallback: feats rows are single-use -> non-temporal loads,
        // software prefetch (global_prefetch_b8) to cover latency in long segments.
        for (; c < len; ++c) {
            const float4v* src = row + (size_t)c * stride4;
            if (c + 2 < len)
                __builtin_prefetch(row + (size_t)(c + 2) * stride4, 0, 0);
            acc += __builtin_nontemporal_load(src);
        }
#endif
        const float inv = 1.0f / (float)len;
        acc *= inv;
    }

    // Always add the position embedding (row 0 is zeros for padding words).
    acc += ((const float4v*)(pos_table + (size_t)p * D_))[t];

    // Output is write-once streaming -> non-temporal store.
    __builtin_nontemporal_store(acc, (float4v*)(out + (size_t)base * D_) + t);
}

extern "C" void kernel_launch(void* const* d_in, const int* in_sizes, int n_in,
                              void* d_out, int out_size, void* d_ws, size_t ws_size,
                              hipStream_t stream) {
    const float* feats     = (const float*)d_in[0];   // [B,S,D] f32
    const int*   word_lens = (const int*)  d_in[1];   // [B,W] i32
    const int*   seq_len   = (const int*)  d_in[2];   // [B] i32
    const int*   pos       = (const int*)  d_in[3];   // [B,W] i32
    const float* pos_table = (const float*)d_in[4];   // [NPOS,D] f32
    float*       out       = (float*)d_out;           // [B,W,D] f32

    (void)in_sizes; (void)n_in; (void)out_size; (void)d_ws; (void)ws_size;

    dim3 grid(W_, B_);   // one block per (word, batch) = 16384 blocks
    dim3 block(TPB);
    seg_mean_pos_kernel<<<grid, block, 0, stream>>>(feats, word_lens, seq_len,
                                                    pos, pos_table, out);
}